// Bottleneck_56298431316575
// MI455X (gfx1250) — compile-verified
//
#include <hip/hip_runtime.h>
#include <hip/hip_bf16.h>

typedef float v2f __attribute__((ext_vector_type(2)));
typedef float v8f __attribute__((ext_vector_type(8)));

#define BATCH 8
#define NV    100000
#define CIN   64
#define DOUT  128
#define KSEL  25000

// D = A(16x4 f32) * B(4x16 f32) + C(16x16 f32)  -- exact fp32 matrix op on CDNA5
__device__ __forceinline__ v8f wmma_f32_16x16x4(v2f a, v2f b, v8f c) {
  return __builtin_amdgcn_wmma_f32_16x16x4_f32(false, a, false, b, (short)0, c, false, false);
}

// Monotone transform: ascending sort on key == descending sort on float value.
__device__ __forceinline__ unsigned key_desc(float f) {
  unsigned u = __float_as_uint(f);
  u = (u & 0x80000000u) ? ~u : (u | 0x80000000u);
  return ~u;
}

// Stage W1 into LDS pre-swizzled so each WMMA B-fragment is one contiguous,
// 8-byte aligned float2:  sW1[((2k+hi)*DOUT + d)*2 + j]  holds W1[(4k+2hi+j)*DOUT + d].
// Fragment load becomes a single ds_load_b64 into an even VGPR pair (no repack movs).
__device__ __forceinline__ void stage_weights(const float* __restrict__ W1,
                                              const float* __restrict__ b1,
                                              const float* __restrict__ W2,
                                              float* sW1, float* sB1, float* sW2)
{
  const int tid = threadIdx.x;
  for (int i = tid; i < CIN * DOUT; i += 256) {
    int c = i >> 7;          // i / DOUT
    int d = i & (DOUT - 1);  // i % DOUT
    int k  = c >> 2;
    int hi = (c >> 1) & 1;
    int j  = c & 1;
    sW1[(((k << 1) | hi) * DOUT + d) * 2 + j] = W1[i];
  }
  if (tid < DOUT) { sB1[tid] = b1[tid]; sW2[tid] = W2[tid]; }
}

// ---------------------------------------------------------------------------
// Kernel 1: fused GEMM (signal @ W1 + b1, relu) -> project W2+b2, relu -> key
// One wave computes a 16-vertex x 128-feature tile fully in WMMA accumulators.
// ---------------------------------------------------------------------------
__global__ void __launch_bounds__(256)
gemm_score_kernel(const float* __restrict__ signal,
                  const float* __restrict__ W1,
                  const float* __restrict__ b1,
                  const float* __restrict__ W2,
                  const float* __restrict__ b2,
                  unsigned* __restrict__ keyA,
                  unsigned* __restrict__ idxA)
{
  __shared__ float sW1[CIN * DOUT];
  __shared__ float sB1[DOUT];
  __shared__ float sW2[DOUT];
  stage_weights(W1, b1, W2, sW1, sB1, sW2);
  __syncthreads();

  const int tid  = threadIdx.x;
  const int wave = tid >> 5;
  const int lane = tid & 31;
  const int m16  = lane & 15;   // row within tile (A) / column (B,C,D)
  const int hi   = lane >> 4;   // half-wave select
  const int b    = blockIdx.y;
  const int tile = blockIdx.x * 8 + wave;
  const int ntiles = NV / 16;   // 6250, exact
  if (tile >= ntiles) return;   // whole-wave uniform exit (after barrier)

  const float* srow = signal + ((size_t)b * NV + (size_t)tile * 16 + m16) * CIN;
  // per-lane swizzled B base for this half-wave / column
  const float* bbase = sW1 + ((size_t)hi * DOUT + m16) * 2;

  v8f acc[8] = {};
  #pragma unroll
  for (int k = 0; k < 16; ++k) {          // K = 64 in steps of 4
    v2f a = *(const v2f*)(srow + 4 * k + 2 * hi);
    const float* bk = bbase + (size_t)(2 * k) * DOUT * 2;
    #pragma unroll
    for (int t = 0; t < 8; ++t) {         // 8 N-tiles of 16 -> 128 features
      v2f bb = *(const v2f*)(bk + 32 * t);   // contiguous pair, ds_load_b64
      acc[t] = wmma_f32_16x16x4(a, bb, acc[t]);
    }
  }

  const float beta = b2[0];
  #pragma unroll
  for (int r = 0; r < 8; ++r) {           // rows r (lanes 0-15) / r+8 (16-31)
    float p = 0.0f;
    #pragma unroll
    for (int t = 0; t < 8; ++t) {
      float y = fmaxf(acc[t][r] + sB1[16 * t + m16], 0.0f);
      p += y * sW2[16 * t + m16];
    }
    // reduce across the 16 columns held by the 16 lanes of each half-wave
    p += __shfl_xor(p, 1, 32);
    p += __shfl_xor(p, 2, 32);
    p += __shfl_xor(p, 4, 32);
    p += __shfl_xor(p, 8, 32);
    if (m16 == 0) {
      float sc = fmaxf(p + beta, 0.0f);
      int n = tile * 16 + r + 8 * hi;
      keyA[(size_t)b * NV + n] = key_desc(sc);
      idxA[(size_t)b * NV + n] = (unsigned)n;
    }
  }
}

// ---------------------------------------------------------------------------
// Kernel 2: per-batch stable LSD radix sort (8 x 4-bit), ascending on key
// => descending score, index-ascending ties (matches lax.top_k ordering).
// One workgroup per batch; per-thread contiguous chunks keep scatter stable.
// ---------------------------------------------------------------------------
__global__ void __launch_bounds__(256)
radix_sort_kernel(unsigned* keyA, unsigned* idxA, unsigned* keyB, unsigned* idxB)
{
  __shared__ unsigned hist[16 * 256];
  const int tid = threadIdx.x;
  const int b   = blockIdx.x;
  const int CH  = (NV + 255) / 256;          // 391
  unsigned* ks = keyA + (size_t)b * NV;
  unsigned* is = idxA + (size_t)b * NV;
  unsigned* kd = keyB + (size_t)b * NV;
  unsigned* id = idxB + (size_t)b * NV;
  const int lo = tid * CH;
  const int hiEnd = (lo + CH < NV) ? (lo + CH) : NV;

  for (int pass = 0; pass < 8; ++pass) {
    const int shift = pass * 4;
    #pragma unroll
    for (int d = 0; d < 16; ++d) hist[d * 256 + tid] = 0;
    __syncthreads();
    for (int i = lo; i < hiEnd; ++i) {
      unsigned dgt = (ks[i] >> shift) & 15u;
      hist[dgt * 256 + tid] += 1;            // thread-private column, no atomics
    }
    __syncthreads();
    if (tid == 0) {                          // digit-major exclusive scan (4096)
      unsigned run = 0;
      for (int j = 0; j < 16 * 256; ++j) {
        unsigned v = hist[j];
        hist[j] = run;
        run += v;
      }
    }
    __syncthreads();
    for (int i = lo; i < hiEnd; ++i) {
      unsigned k = ks[i];
      unsigned dgt = (k >> shift) & 15u;
      unsigned pos = hist[dgt * 256 + tid]++;
      kd[pos] = k;
      id[pos] = is[i];
    }
    __syncthreads();
    unsigned* t;
    t = ks; ks = kd; kd = t;
    t = is; is = id; id = t;
  }
  // 8 (even) passes: final data back in keyA/idxA
}

// ---------------------------------------------------------------------------
// Kernel 3: gather top-K coordinates -> d_out[0 : B*K*3]
// ---------------------------------------------------------------------------
__global__ void gather_coords_kernel(const float* __restrict__ coords,
                                     const unsigned* __restrict__ sortedIdx,
                                     float* __restrict__ outCoords)
{
  int gid = blockIdx.x * blockDim.x + threadIdx.x;
  const int total = BATCH * KSEL;
  if (gid >= total) return;
  int b  = gid / KSEL;
  int kk = gid - b * KSEL;
  unsigned n = sortedIdx[(size_t)b * NV + kk];
  const float* src = coords + ((size_t)b * NV + n) * 3;
  float* dst = outCoords + (size_t)gid * 3;
  dst[0] = src[0]; dst[1] = src[1]; dst[2] = src[2];
}

// ---------------------------------------------------------------------------
// Kernel 4: recompute features via WMMA for selected vertices and write
// gathered [B, K, 129] (= concat(relu(sig@W1+b1), relu(.@W2+b2))).
// ---------------------------------------------------------------------------
__global__ void __launch_bounds__(256)
gather_feat_kernel(const float* __restrict__ signal,
                   const float* __restrict__ W1,
                   const float* __restrict__ b1,
                   const float* __restrict__ W2,
                   const float* __restrict__ b2,
                   const unsigned* __restrict__ sortedIdx,
                   float* __restrict__ outSig)
{
  __shared__ float sW1[CIN * DOUT];
  __shared__ float sB1[DOUT];
  __shared__ float sW2[DOUT];
  stage_weights(W1, b1, W2, sW1, sB1, sW2);
  __syncthreads();

  const int tid  = threadIdx.x;
  const int wave = tid >> 5;
  const int lane = tid & 31;
  const int m16  = lane & 15;
  const int hi   = lane >> 4;
  const int b    = blockIdx.y;
  const int tile = blockIdx.x * 8 + wave;
  const int ntiles = (KSEL + 15) / 16;   // 1563 (last tile partial)
  if (tile >= ntiles) return;

  int kkLoad = tile * 16 + m16;
  if (kkLoad > KSEL - 1) kkLoad = KSEL - 1;   // clamp loads, keep EXEC all-1s
  const unsigned n = sortedIdx[(size_t)b * NV + kkLoad];
  const float* srow  = signal + ((size_t)b * NV + n) * CIN;
  const float* bbase = sW1 + ((size_t)hi * DOUT + m16) * 2;

  v8f acc[8] = {};
  #pragma unroll
  for (int k = 0; k < 16; ++k) {
    v2f a = *(const v2f*)(srow + 4 * k + 2 * hi);
    const float* bk = bbase + (size_t)(2 * k) * DOUT * 2;
    #pragma unroll
    for (int t = 0; t < 8; ++t) {
      v2f bb = *(const v2f*)(bk + 32 * t);
      acc[t] = wmma_f32_16x16x4(a, bb, acc[t]);
    }
  }

  const float beta = b2[0];
  #pragma unroll
  for (int r = 0; r < 8; ++r) {
    float p = 0.0f;
    #pragma unroll
    for (int t = 0; t < 8; ++t) {
      float y = fmaxf(acc[t][r] + sB1[16 * t + m16], 0.0f);
      acc[t][r] = y;
      p += y * sW2[16 * t + m16];
    }
    p += __shfl_xor(p, 1, 32);
    p += __shfl_xor(p, 2, 32);
    p += __shfl_xor(p, 4, 32);
    p += __shfl_xor(p, 8, 32);
    const int kkRow = tile * 16 + r + 8 * hi;   // row this lane's acc holds
    if (kkRow < KSEL) {
      float* orow = outSig + ((size_t)b * KSEL + kkRow) * (DOUT + 1);
      #pragma unroll
      for (int t = 0; t < 8; ++t)
        orow[16 * t + m16] = acc[t][r];
      if (m16 == 0)
        orow[DOUT] = fmaxf(p + beta, 0.0f);     // the "down" score channel
    }
  }
}

// ---------------------------------------------------------------------------
extern "C" void kernel_launch(void* const* d_in, const int* in_sizes, int n_in,
                              void* d_out, int out_size, void* d_ws, size_t ws_size,
                              hipStream_t stream)
{
  const float* coords = (const float*)d_in[0];   // [B,N,3]
  const float* signal = (const float*)d_in[1];   // [B,N,C]
  const float* W1     = (const float*)d_in[2];   // [C,D]
  const float* b1     = (const float*)d_in[3];   // [D]
  const float* W2     = (const float*)d_in[4];   // [D,1]
  const float* b2     = (const float*)d_in[5];   // [1]
  float* out = (float*)d_out;

  // Workspace: sort ping-pong buffers (4 x B*N u32 = 12.8 MB total)
  unsigned* keyA = (unsigned*)d_ws;
  unsigned* idxA = keyA + (size_t)BATCH * NV;
  unsigned* keyB = idxA + (size_t)BATCH * NV;
  unsigned* idxB = keyB + (size_t)BATCH * NV;

  dim3 grid1((NV / 16 + 7) / 8, BATCH);          // 782 x 8, 8 tiles/block
  gemm_score_kernel<<<grid1, 256, 0, stream>>>(signal, W1, b1, W2, b2, keyA, idxA);

  radix_sort_kernel<<<BATCH, 256, 0, stream>>>(keyA, idxA, keyB, idxB);

  const int totalC = BATCH * KSEL;
  gather_coords_kernel<<<(totalC + 255) / 256, 256, 0, stream>>>(coords, idxA, out);

  float* outSig = out + (size_t)BATCH * KSEL * 3;
  dim3 grid4((((KSEL + 15) / 16) + 7) / 8, BATCH);   // 196 x 8
  gather_feat_kernel<<<grid4, 256, 0, stream>>>(signal, W1, b1, W2, b2, idxA, outSig);
}